// GATv2_25134148617002
// MI455X (gfx1250) — compile-verified
//
#include <hip/hip_runtime.h>
#include <hip/hip_bf16.h>

// GATv2 for MI455X (gfx1250).
//  K1: Q/K node GEMMs via V_WMMA_F32_16X16X4_F32, weights staged in LDS.
//      One wave = one 16x128 strip of ONE output matrix (8 v8f accumulators,
//      64 VGPRs -> no spills; round-1 version with 16 accumulators spilled).
//  K2: edge logits + segment-max via ordered-uint atomicMax (L2-resident gathers)
//  K3: edge exp + segment-sum via atomicAdd
//  K4: edge scatter: k * coeff atomic-added into d_out
//  K5: in-place ReLU
// Edge phase is L2-gather/atomic bound; q,k (51 MB) fit in the 192 MB L2.

typedef float v2f __attribute__((ext_vector_type(2)));
typedef float v8f __attribute__((ext_vector_type(8)));

#define D_IN      128
#define N_HEADS   8
#define PER_HEAD  16
#define LEAKY     0.2f

// ---------------------------------------------------------------- init ------
__global__ void gat_zero_kernel(float* __restrict__ out, size_t n_out,
                                unsigned* __restrict__ segmax, float* __restrict__ denom,
                                size_t n_seg) {
    size_t i = (size_t)blockIdx.x * blockDim.x + threadIdx.x;
    size_t stride = (size_t)gridDim.x * blockDim.x;
    for (size_t j = i; j < n_out; j += stride) out[j] = 0.0f;
    for (size_t j = i; j < n_seg; j += stride) { segmax[j] = 0u; denom[j] = 0.0f; }
}

// ------------------------------------------------ K1: Q/K GEMM via WMMA -----
// blockIdx.y selects which matrix (0 -> query/Wq, 1 -> key/Wk).
// Block = 4 waves; W (128x128 fp32 = 64 KB) is cooperatively staged into LDS,
// interleaved by K-row pairs so each B fragment is one ds_load_b64:
//   ldsW[(k>>1)*256 + col*2 + (k&1)] = W[k][col]
__global__ __launch_bounds__(128)
void gat_qk_gemm_kernel(const float* __restrict__ x,
                        const float* __restrict__ wq, const float* __restrict__ bq,
                        const float* __restrict__ wk, const float* __restrict__ bk,
                        float* __restrict__ q, float* __restrict__ k,
                        int n_tiles) {
    __shared__ float ldsW[D_IN * D_IN];          // 64 KB

    const float* __restrict__ w    = blockIdx.y ? wk : wq;
    const float* __restrict__ bias = blockIdx.y ? bk : bq;
    float* __restrict__ outm       = blockIdx.y ? k  : q;

    // ---- stage W into LDS (interleaved K-row pairs), all threads ----
    for (int i = threadIdx.x; i < D_IN * D_IN; i += blockDim.x) {
        const int kr  = i >> 7;        // K row
        const int col = i & 127;       // output column
        ldsW[(kr >> 1) * 256 + col * 2 + (kr & 1)] = w[i];
    }
    __syncthreads();

    const int lane = threadIdx.x & 31;
    const int wave = threadIdx.x >> 5;
    const int tile = blockIdx.x * 4 + wave;
    if (tile < n_tiles) {
        const int mlo  = lane & 15;        // M (A) / N (B,C) index within tile
        const int koff = (lane >> 4) * 2;  // K sub-offset: lanes 0-15 -> K=0,1 ; 16-31 -> K=2,3
        const float* __restrict__ xrow = x + (size_t)(tile * 16 + mlo) * D_IN;

        v8f acc[8];
        const v8f vzero = {0.f, 0.f, 0.f, 0.f, 0.f, 0.f, 0.f, 0.f};
#pragma unroll
        for (int t = 0; t < 8; ++t) acc[t] = vzero;

        for (int kb = 0; kb < D_IN; kb += 4) {
            v2f a;                                       // A: 16x4 fp32 fragment
            a.x = xrow[kb + koff + 0];
            a.y = xrow[kb + koff + 1];
            const float* __restrict__ wrow = ldsW + ((kb + koff) >> 1) * 256;
#pragma unroll
            for (int t = 0; t < 8; ++t) {                // 8 column tiles of 16
                const int col = t * 16 + mlo;
                v2f b = *(const v2f*)(wrow + col * 2);   // B: 4x16 fp32 fragment (b64)
                acc[t] = __builtin_amdgcn_wmma_f32_16x16x4_f32(
                    false, a, false, b, (short)0, acc[t], false, false);
            }
        }

        // C/D layout: VGPR r -> lanes 0-15: (M=r, N=lane), lanes 16-31: (M=r+8, N=lane-16)
        const int mhi = (lane >> 4) * 8;
#pragma unroll
        for (int t = 0; t < 8; ++t) {
            const int col = t * 16 + mlo;
            const float bs = bias[col];
#pragma unroll
            for (int r = 0; r < 8; ++r) {
                const size_t row = (size_t)(tile * 16 + mhi + r);
                outm[row * D_IN + col] = acc[t][r] + bs;
            }
        }
    }
}

// ------------------------------------------- fp32 <-> order-preserving uint --
__device__ __forceinline__ unsigned f32_ord(float f) {
    unsigned u = __float_as_uint(f);
    return (u & 0x80000000u) ? ~u : (u | 0x80000000u);
}
__device__ __forceinline__ float ord_f32(unsigned u) {
    unsigned v = (u & 0x80000000u) ? (u & 0x7FFFFFFFu) : ~u;
    return __uint_as_float(v);
}

__device__ __forceinline__ float lrelu(float v) {
    return fmaxf(v, 0.0f) + LEAKY * fminf(v, 0.0f);
}

// ----------------------------- K2: per-(edge,head) logits + segment max -----
__global__ void gat_edge_logits_kernel(const float* __restrict__ q,
                                       const float* __restrict__ k,
                                       const float* __restrict__ attn,   // [16, 8]
                                       const int* __restrict__ src,
                                       const int* __restrict__ dst,
                                       float* __restrict__ logits,       // [E, 8]
                                       unsigned* __restrict__ segmax,    // [N, 8]
                                       long long n_work) {
    long long idx = (long long)blockIdx.x * blockDim.x + threadIdx.x;
    if (idx >= n_work) return;
    const int e = (int)(idx >> 3);
    const int h = (int)(idx & 7);
    const int s = src[e];
    const int t = dst[e];
    const float4* __restrict__ qp = (const float4*)(q + (size_t)t * D_IN + h * PER_HEAD);
    const float4* __restrict__ kp = (const float4*)(k + (size_t)s * D_IN + h * PER_HEAD);
    float acc = 0.0f;
#pragma unroll
    for (int i = 0; i < 4; ++i) {
        float4 qv = qp[i];
        float4 kv = kp[i];
        acc += lrelu(qv.x + kv.x) * attn[(i * 4 + 0) * N_HEADS + h];
        acc += lrelu(qv.y + kv.y) * attn[(i * 4 + 1) * N_HEADS + h];
        acc += lrelu(qv.z + kv.z) * attn[(i * 4 + 2) * N_HEADS + h];
        acc += lrelu(qv.w + kv.w) * attn[(i * 4 + 3) * N_HEADS + h];
    }
    logits[idx] = acc;
    atomicMax(segmax + (size_t)t * N_HEADS + h, f32_ord(acc));
}

// --------------------------- K3: exp(logit - segmax) + segment sum ----------
__global__ void gat_edge_expsum_kernel(const int* __restrict__ dst,
                                       const unsigned* __restrict__ segmax,
                                       float* __restrict__ logits,   // in: logits, out: unnorm
                                       float* __restrict__ denom,    // [N, 8]
                                       long long n_work) {
    long long idx = (long long)blockIdx.x * blockDim.x + threadIdx.x;
    if (idx >= n_work) return;
    const int e = (int)(idx >> 3);
    const int h = (int)(idx & 7);
    const int t = dst[e];
    const float m = ord_f32(segmax[(size_t)t * N_HEADS + h]);
    const float u = __expf(logits[idx] - m);
    logits[idx] = u;
    atomicAdd(denom + (size_t)t * N_HEADS + h, u);
}

// ----------------- K4: messages = k * coeff, scatter-add into pooled --------
__global__ void gat_edge_scatter_kernel(const float* __restrict__ k,
                                        const int* __restrict__ src,
                                        const int* __restrict__ dst,
                                        const float* __restrict__ unnorm,  // [E, 8]
                                        const float* __restrict__ denom,   // [N, 8]
                                        float* __restrict__ pooled,        // [N, 128] (= d_out)
                                        long long n_work) {
    long long idx = (long long)blockIdx.x * blockDim.x + threadIdx.x;
    if (idx >= n_work) return;
    const int e = (int)(idx >> 3);
    const int h = (int)(idx & 7);
    const int s = src[e];
    const int t = dst[e];
    const float coeff = unnorm[idx] / denom[(size_t)t * N_HEADS + h];
    const float4* __restrict__ kp = (const float4*)(k + (size_t)s * D_IN + h * PER_HEAD);
    float* __restrict__ op = pooled + (size_t)t * D_IN + h * PER_HEAD;
#pragma unroll
    for (int i = 0; i < 4; ++i) {
        float4 kv = kp[i];
        atomicAdd(op + i * 4 + 0, kv.x * coeff);
        atomicAdd(op + i * 4 + 1, kv.y * coeff);
        atomicAdd(op + i * 4 + 2, kv.z * coeff);
        atomicAdd(op + i * 4 + 3, kv.w * coeff);
    }
}

// ----------------------------------------------- K5: in-place ReLU ----------
__global__ void gat_relu_kernel(float* __restrict__ out, size_t n) {
    size_t i = (size_t)blockIdx.x * blockDim.x + threadIdx.x;
    if (i < n) out[i] = fmaxf(out[i], 0.0f);
}

// ---------------------------------------------------------------------------
extern "C" void kernel_launch(void* const* d_in, const int* in_sizes, int n_in,
                              void* d_out, int out_size, void* d_ws, size_t ws_size,
                              hipStream_t stream) {
    const float* x    = (const float*)d_in[0];
    const float* wq   = (const float*)d_in[1];
    const float* bq   = (const float*)d_in[2];
    const float* wk   = (const float*)d_in[3];
    const float* bk   = (const float*)d_in[4];
    const float* attn = (const float*)d_in[5];
    const int*   esrc = (const int*)d_in[6];
    const int*   edst = (const int*)d_in[7];

    const int n_nodes = in_sizes[0] / D_IN;     // 50000
    const int n_edges = in_sizes[6];            // 800000

    // Workspace layout (fp32 unless noted):
    //   q[N*128] | k[N*128] | logits[E*8] | segmax[N*8] (uint) | denom[N*8]
    float*    q      = (float*)d_ws;
    float*    k      = q + (size_t)n_nodes * D_IN;
    float*    logits = k + (size_t)n_nodes * D_IN;
    unsigned* segmax = (unsigned*)(logits + (size_t)n_edges * N_HEADS);
    float*    denom  = (float*)(segmax + (size_t)n_nodes * N_HEADS);

    float* out = (float*)d_out;

    // K0: zero pooled accumulator (d_out), segmax sentinel, denom.
    {
        const size_t n_out = (size_t)n_nodes * D_IN;
        const size_t n_seg = (size_t)n_nodes * N_HEADS;
        gat_zero_kernel<<<2048, 256, 0, stream>>>(out, n_out, segmax, denom, n_seg);
    }

    // K1: Q/K GEMMs with fp32 WMMA. grid.y = {Q, K}; 4 waves/block, 1 tile/wave.
    {
        const int n_tiles = n_nodes / 16;                 // 3125 (50000 % 16 == 0)
        dim3 grid((n_tiles + 3) / 4, 2);
        gat_qk_gemm_kernel<<<grid, 128, 0, stream>>>(x, wq, bq, wk, bk, q, k, n_tiles);
    }

    const long long n_work = (long long)n_edges * N_HEADS;   // 6.4M
    const int eb = 256;
    const int egrid = (int)((n_work + eb - 1) / eb);

    // K2: logits + segment max
    gat_edge_logits_kernel<<<egrid, eb, 0, stream>>>(q, k, attn, esrc, edst,
                                                     logits, segmax, n_work);
    // K3: exp + segment sum
    gat_edge_expsum_kernel<<<egrid, eb, 0, stream>>>(edst, segmax, logits, denom, n_work);
    // K4: weighted scatter into d_out
    gat_edge_scatter_kernel<<<egrid, eb, 0, stream>>>(k, esrc, edst, logits, denom,
                                                      out, n_work);
    // K5: ReLU in place
    {
        const size_t n_out = (size_t)n_nodes * D_IN;
        const int blocks = (int)((n_out + 255) / 256);
        gat_relu_kernel<<<blocks, 256, 0, stream>>>(out, n_out);
    }
}